// _VarGatePN_65481071404294
// MI455X (gfx1250) — compile-verified
//
#include <hip/hip_runtime.h>
#include <math.h>

// ---------------------------------------------------------------------------
// Problem: angles = x @ W.T  (16384x256 * 256x10, fp32), then a 10-qubit
// statevector simulation per batch row -> expvals [16384, 10] fp32.
//
// Kernel 1: fp32 WMMA GEMM (V_WMMA_F32_16X16X4_F32), one wave per 16-row tile.
//           W.T staged zero-padded in LDS so the inner loop has no divergent
//           loads (WMMA needs EXEC all-1s; exec-masked loads per iteration
//           were serializing round 1's loop).
// Kernel 2: one wave per batch element; 1024-complex state held in VGPRs
//           (32 complex per lane), cross-lane gates via __shfl_xor.
// ---------------------------------------------------------------------------

typedef __attribute__((ext_vector_type(2))) float v2f;
typedef __attribute__((ext_vector_type(8))) float v8f;

#define N_WIRES 10
#define DEPTH   2
#define K_DIM   256

// ------------------------------ GEMM ---------------------------------------
// A (16x4 f32): lanes 0-15 hold row M=lane, K=0 (v0), K=1 (v1);
//               lanes 16-31 hold row M=lane-16, K=2 (v0), K=3 (v1).
// B (4x16 f32): v0 = rows K=0 (lanes 0-15) / K=2 (lanes 16-31), v1 = K=1/K=3.
// D (16x16 f32): VGPR j -> M=j (lanes 0-15), M=j+8 (lanes 16-31), N=lane&15.
__global__ __launch_bounds__(256)
void gemm_angles_wmma(const float* __restrict__ x, const float* __restrict__ W,
                      float* __restrict__ angles) {
  __shared__ float wlds[16 * K_DIM];          // W zero-padded to 16 rows, 16 KB

  // Cooperative load + zero-pad (rows 10..15 = 0). 4096 floats / 256 threads.
  for (int i = threadIdx.x; i < 16 * K_DIM; i += 256) {
    const int n = i >> 8;                     // row (output wire)
    wlds[i] = (n < N_WIRES) ? W[i] : 0.f;
  }
  __syncthreads();

  const int wave = (blockIdx.x * blockDim.x + threadIdx.x) >> 5;   // tile id
  const int lane = threadIdx.x & 31;
  const int half = lane >> 4;        // 0: K+0/K+1, 1: K+2/K+3
  const int l    = lane & 15;
  const int m0   = wave * 16;

  const float* __restrict__ xrow = x + (size_t)(m0 + l) * K_DIM + half * 2;
  const float* __restrict__ wrow = &wlds[l * K_DIM + half * 2];

  // Two independent accumulator chains to break the WMMA RAW dependency.
  v8f acc0 = {0.f, 0.f, 0.f, 0.f, 0.f, 0.f, 0.f, 0.f};
  v8f acc1 = {0.f, 0.f, 0.f, 0.f, 0.f, 0.f, 0.f, 0.f};

#pragma unroll 4
  for (int k = 0; k < K_DIM; k += 8) {
    v2f a0, b0, a1, b1;
    a0.x = xrow[k];     a0.y = xrow[k + 1];
    b0.x = wrow[k];     b0.y = wrow[k + 1];
    a1.x = xrow[k + 4]; a1.y = xrow[k + 5];
    b1.x = wrow[k + 4]; b1.y = wrow[k + 5];
    acc0 = __builtin_amdgcn_wmma_f32_16x16x4_f32(
        false, a0, false, b0, (short)0, acc0, false, false);
    acc1 = __builtin_amdgcn_wmma_f32_16x16x4_f32(
        false, a1, false, b1, (short)0, acc1, false, false);
  }

#pragma unroll
  for (int j = 0; j < 8; ++j) {
    const int row = m0 + j + half * 8;
    angles[(size_t)row * 16 + l] = acc0[j] + acc1[j];
  }
}

// --------------------------- quantum sim -----------------------------------
// State index i in [0,1024): bits[9:5] = lane, bits[4:0] = register r.
// Wire w acts on index bit p = 9 - w  (wire 0 is MSB, Pennylane convention).

__device__ __forceinline__ void rx_gate(float (&re)[32], float (&im)[32],
                                        int w, float c, float s) {
  const int p = 9 - w;
  if (p >= 5) {                       // cross-lane: partner = lane ^ (1<<(p-5))
    const int m = 1 << (p - 5);
#pragma unroll
    for (int r = 0; r < 32; ++r) {
      const float pre = __shfl_xor(re[r], m, 32);
      const float pim = __shfl_xor(im[r], m, 32);
      // new = c*mine + s*(partner.im, -partner.re)  (valid for both halves)
      const float nre = c * re[r] + s * pim;
      const float nim = c * im[r] - s * pre;
      re[r] = nre;
      im[r] = nim;
    }
  } else {                            // in-register pairs (r, r+st)
    const int st = 1 << p;
#pragma unroll
    for (int r = 0; r < 32; ++r) {
      if ((r & st) == 0) {
        const int r1 = r + st;
        const float a0r = re[r],  a0i = im[r];
        const float a1r = re[r1], a1i = im[r1];
        re[r]  = c * a0r + s * a1i;
        im[r]  = c * a0i - s * a1r;
        re[r1] = c * a1r + s * a0i;
        im[r1] = c * a1i - s * a0r;
      }
    }
  }
}

__device__ __forceinline__ void cnot_gate(float (&re)[32], float (&im)[32],
                                          int w, int lane) {
  const int cb = 9 - w;   // control bit position
  const int tb = 8 - w;   // target bit position
  if (tb >= 5) {          // control & target both lane bits (w = 0..3)
    const int m = 1 << (tb - 5);
    const bool ctrl = (lane >> (cb - 5)) & 1;
#pragma unroll
    for (int r = 0; r < 32; ++r) {
      const float tre = __shfl_xor(re[r], m, 32);
      const float tim = __shfl_xor(im[r], m, 32);
      re[r] = ctrl ? tre : re[r];
      im[r] = ctrl ? tim : im[r];
    }
  } else if (cb >= 5) {   // w == 4: lane-bit control, register-bit target
    const bool ctrl = (lane >> (cb - 5)) & 1;
    const int st = 1 << tb;
#pragma unroll
    for (int r = 0; r < 32; ++r) {
      if ((r & st) == 0) {
        const int r1 = r + st;
        float t;
        t = re[r];  re[r]  = ctrl ? re[r1] : re[r];  re[r1] = ctrl ? t : re[r1];
        t = im[r];  im[r]  = ctrl ? im[r1] : im[r];  im[r1] = ctrl ? t : im[r1];
      }
    }
  } else {                // both register bits (w >= 5): pure register swap
    const int st = 1 << tb;
    const int cm = 1 << cb;
#pragma unroll
    for (int r = 0; r < 32; ++r) {
      if (((r & cm) != 0) && ((r & st) == 0)) {
        const int r1 = r + st;
        float t;
        t = re[r]; re[r] = re[r1]; re[r1] = t;
        t = im[r]; im[r] = im[r1]; im[r1] = t;
      }
    }
  }
}

__global__ __launch_bounds__(256)
void qsim_kernel(const float* __restrict__ angles,   // [B,16] padded
                 const float* __restrict__ params,   // [DEPTH, N_WIRES]
                 float* __restrict__ out) {          // [B, N_WIRES]
  const int lane = threadIdx.x & 31;
  const int b    = (blockIdx.x * blockDim.x + threadIdx.x) >> 5;

  float re[32], im[32];
#pragma unroll
  for (int r = 0; r < 32; ++r) { re[r] = 0.f; im[r] = 0.f; }
  re[0] = (lane == 0) ? 1.f : 0.f;   // |0...0>

  // lane k (k<16) holds angle for wire k; broadcast with __shfl.
  const float myang = angles[(size_t)b * 16 + (lane & 15)];

  // Layer 0: data-dependent RX per wire.
#pragma unroll
  for (int w = 0; w < N_WIRES; ++w) {
    const float th = __shfl(myang, w, 32) * 0.5f;
    rx_gate(re, im, w, cosf(th), sinf(th));
  }

  // Variational layers.
#pragma unroll
  for (int d = 0; d < DEPTH; ++d) {
#pragma unroll
    for (int w = 0; w < N_WIRES; ++w) {
      const float th = params[d * N_WIRES + w] * 0.5f;
      rx_gate(re, im, w, cosf(th), sinf(th));
    }
#pragma unroll
    for (int w = 0; w < N_WIRES - 1; ++w) {
      cnot_gate(re, im, w, lane);
    }
  }

  // Probabilities and per-wire expvals.
  float p[32];
  float ptot = 0.f;
#pragma unroll
  for (int r = 0; r < 32; ++r) {
    p[r] = re[r] * re[r] + im[r] * im[r];
    ptot += p[r];
  }

  float e[N_WIRES];
#pragma unroll
  for (int w = 0; w < 5; ++w) {              // bit 9-w is a lane bit (4-w)
    e[w] = ((lane >> (4 - w)) & 1) ? -ptot : ptot;
  }
#pragma unroll
  for (int w = 5; w < N_WIRES; ++w) {        // bit 9-w is a register bit
    const int bp = 9 - w;
    float acc = 0.f;
#pragma unroll
    for (int r = 0; r < 32; ++r) acc += ((r >> bp) & 1) ? -p[r] : p[r];
    e[w] = acc;
  }

  // Butterfly reduce each expval across the wave.
#pragma unroll
  for (int w = 0; w < N_WIRES; ++w) {
#pragma unroll
    for (int m = 1; m < 32; m <<= 1) e[w] += __shfl_xor(e[w], m, 32);
  }

  if (lane == 0) {
#pragma unroll
    for (int w = 0; w < N_WIRES; ++w) out[(size_t)b * N_WIRES + w] = e[w];
  }
}

// ------------------------------ launcher -----------------------------------
extern "C" void kernel_launch(void* const* d_in, const int* in_sizes, int n_in,
                              void* d_out, int out_size, void* d_ws, size_t ws_size,
                              hipStream_t stream) {
  const float* x      = (const float*)d_in[0];  // [B, 256]
  const float* W      = (const float*)d_in[1];  // [10, 256]
  const float* params = (const float*)d_in[2];  // [2, 10, 1]
  float* out    = (float*)d_out;                // [B, 10]
  float* angles = (float*)d_ws;                 // [B, 16] scratch (B*64 bytes)

  const int B = in_sizes[0] / K_DIM;            // 16384

  // GEMM: one wave per 16 rows -> B/16 waves, 8 waves (256 thr) per block.
  const int gemm_blocks = (B / 16 + 7) / 8;
  hipLaunchKernelGGL(gemm_angles_wmma, dim3(gemm_blocks), dim3(256), 0, stream,
                     x, W, angles);

  // Sim: one wave per batch element -> B waves, 8 per block.
  const int sim_blocks = (B + 7) / 8;
  hipLaunchKernelGGL(qsim_kernel, dim3(sim_blocks), dim3(256), 0, stream,
                     angles, params, out);
}